// BIG_8993661518238
// MI455X (gfx1250) — compile-verified
//
#include <hip/hip_runtime.h>
#include <hip/hip_bf16.h>

#define BN_EPS 1e-5f
#define HID 64

typedef __attribute__((ext_vector_type(2))) float v2f;
typedef __attribute__((ext_vector_type(8))) float v8f;

// ---------------------------------------------------------------------------
// Degree / normalization
// ---------------------------------------------------------------------------
__global__ void deg_init_kernel(float* __restrict__ deg, int n) {
    int i = blockIdx.x * blockDim.x + threadIdx.x;
    if (i < n) deg[i] = 1.0f;  // self loop
}

__global__ void deg_edges_kernel(const long long* __restrict__ dst,
                                 float* __restrict__ deg, int e) {
    int i = blockIdx.x * blockDim.x + threadIdx.x;
    if (i < e) atomicAdd(&deg[(int)dst[i]], 1.0f);
}

__global__ void deg_finish_kernel(float* __restrict__ deg, int n) {
    int i = blockIdx.x * blockDim.x + threadIdx.x;
    if (i < n) deg[i] = rsqrtf(deg[i]);  // deg >= 1 always -> dinv
}

// per-edge weight w = dinv[src]*dinv[dst] (identical for every layer)
__global__ void edge_w_kernel(const long long* __restrict__ src,
                              const long long* __restrict__ dst,
                              const float* __restrict__ dinv,
                              float* __restrict__ wedge, int e) {
    int i = blockIdx.x * blockDim.x + threadIdx.x;
    if (i < e) wedge[i] = dinv[(int)src[i]] * dinv[(int)dst[i]];
}

// ---------------------------------------------------------------------------
// BatchNorm column statistics: stats[0..C) = sum, stats[128..128+C) = sumsq
// ---------------------------------------------------------------------------
__global__ void stats_zero_kernel(float* __restrict__ stats) {
    stats[threadIdx.x] = 0.0f;  // 256 threads, 1 block
}

__global__ __launch_bounds__(256) void stats_kernel(const float* __restrict__ in,
                                                    float* __restrict__ stats,
                                                    int n, int C) {
    __shared__ float s[128];
    __shared__ float sq[128];
    for (int i = threadIdx.x; i < C; i += 256) { s[i] = 0.0f; sq[i] = 0.0f; }
    __syncthreads();
    size_t total  = (size_t)n * (size_t)C;
    size_t stride = (size_t)gridDim.x * 256u;
    for (size_t idx = (size_t)blockIdx.x * 256u + threadIdx.x; idx < total; idx += stride) {
        float v = in[idx];
        int   c = (int)(idx & (size_t)(C - 1));  // C is a power of two
        atomicAdd(&s[c],  v);      // ds_add_f32
        atomicAdd(&sq[c], v * v);
    }
    __syncthreads();
    for (int i = threadIdx.x; i < C; i += 256) {
        atomicAdd(&stats[i],       s[i]);
        atomicAdd(&stats[128 + i], sq[i]);
    }
}

// Fused: h = relu(agg), plus column stats of h for the next layer's BN.
__global__ __launch_bounds__(256) void relu_stats_kernel(const float* __restrict__ agg,
                                                         float* __restrict__ h,
                                                         float* __restrict__ stats, int n) {
    __shared__ float s[64];
    __shared__ float sq[64];
    for (int i = threadIdx.x; i < 64; i += 256) { s[i] = 0.0f; sq[i] = 0.0f; }
    __syncthreads();
    size_t total  = (size_t)n * 64u;
    size_t stride = (size_t)gridDim.x * 256u;
    for (size_t idx = (size_t)blockIdx.x * 256u + threadIdx.x; idx < total; idx += stride) {
        float v = fmaxf(agg[idx], 0.0f);
        h[idx]  = v;
        int c = (int)(idx & 63u);
        atomicAdd(&s[c],  v);
        atomicAdd(&sq[c], v * v);
    }
    __syncthreads();
    for (int i = threadIdx.x; i < 64; i += 256) {
        atomicAdd(&stats[i],       s[i]);
        atomicAdd(&stats[128 + i], sq[i]);
    }
}

__global__ void relu_kernel(const float* __restrict__ in, float* __restrict__ out, int n64) {
    int idx = blockIdx.x * blockDim.x + threadIdx.x;
    if (idx < n64) out[idx] = fmaxf(in[idx], 0.0f);
}

// ---------------------------------------------------------------------------
// Fold BN affine into GEMM weights:
//   a[k] = gamma[k]*rsqrt(var+eps), c[k] = beta[k] - mean[k]*a[k]
//   Wp[k][j] = a[k]*W[k][j];  bp[j] = (b?b[j]:0) + sum_k c[k]*W[k][j]
// ---------------------------------------------------------------------------
__global__ __launch_bounds__(256) void prep_kernel(const float* __restrict__ stats,
                                                   const float* __restrict__ gamma,
                                                   const float* __restrict__ beta,
                                                   const float* __restrict__ Wsrc,
                                                   const float* __restrict__ bsrc,
                                                   float* __restrict__ Wp,
                                                   float* __restrict__ bp,
                                                   int C, float invN) {
    __shared__ float a[128];
    __shared__ float c[128];
    int tid = threadIdx.x;
    for (int k = tid; k < C; k += 256) {
        float mean = stats[k] * invN;
        float var  = stats[128 + k] * invN - mean * mean;
        float av   = gamma[k] * rsqrtf(var + BN_EPS);
        a[k] = av;
        c[k] = beta[k] - mean * av;
    }
    __syncthreads();
    for (int i = tid; i < C * HID; i += 256) {
        int k = i >> 6;  // row (K index)
        Wp[i] = a[k] * Wsrc[i];
    }
    if (tid < HID) {
        float sum = bsrc ? bsrc[tid] : 0.0f;
        for (int k = 0; k < C; ++k) sum += c[k] * Wsrc[k * HID + tid];
        bp[tid] = sum;
    }
}

// ---------------------------------------------------------------------------
// WMMA fp32 GEMM: Out[M x 64] = X[M x K] @ Wp[K x 64] + bp.
// 256 threads = 8 waves; each wave owns one 16x16 tile of a 32x64 block tile.
// Epilogue variants:
//   Agg == nullptr : Out = relu(v) if doRelu else v
//   Agg != nullptr : Out = v (the pre-aggregation t) AND
//                    Agg = bias2 + v * dinv[row]^2   (fused self-loop init)
// ---------------------------------------------------------------------------
__global__ __launch_bounds__(256) void gemm_wmma_kernel(const float* __restrict__ X,
                                                        const float* __restrict__ Wp,
                                                        const float* __restrict__ bp,
                                                        float* __restrict__ Out,
                                                        const float* __restrict__ dinv,
                                                        const float* __restrict__ bias2,
                                                        float* __restrict__ Agg,
                                                        int n, int K, int doRelu) {
    __shared__ float As[32 * 128];   // A tile (32 rows x K), dense K-major
    __shared__ float Bs[128 * HID];  // full Wp (K x 64)

    const int tid     = threadIdx.x;
    const int rowBase = blockIdx.x * 32;

    // --- cooperative loads (float4, fully coalesced) ---
    const int kv4 = K >> 2;  // float4s per A row
    for (int i = tid; i < 32 * kv4; i += 256) {
        int r  = i / kv4;
        int c4 = i - r * kv4;
        int gr = rowBase + r; if (gr >= n) gr = n - 1;  // clamp (n % 32 == 0 normally)
        ((float4*)As)[i] = ((const float4*)(X + (size_t)gr * K))[c4];
    }
    for (int i = tid; i < (K * HID) >> 2; i += 256) {
        ((float4*)Bs)[i] = ((const float4*)Wp)[i];
    }
    __syncthreads();

    const int wave = tid >> 5;     // 0..7
    const int lane = tid & 31;
    const int half = lane >> 4;    // lane group: 0 or 1
    const int l16  = lane & 15;
    const int rt   = wave >> 2;    // row tile 0..1
    const int ct   = wave & 3;     // col tile 0..3

    const float* Arow = As + (rt * 16 + l16) * K;  // this lane's A row
    v8f acc = {0.f, 0.f, 0.f, 0.f, 0.f, 0.f, 0.f, 0.f};

    for (int kk = 0; kk < K; kk += 4) {
        const int kb = kk + half * 2;
        v2f a; a[0] = Arow[kb];                        a[1] = Arow[kb + 1];
        v2f b; b[0] = Bs[kb * HID + ct * 16 + l16];    b[1] = Bs[(kb + 1) * HID + ct * 16 + l16];
        acc = __builtin_amdgcn_wmma_f32_16x16x4_f32(
            /*neg_a=*/false, a, /*neg_b=*/false, b,
            /*c_mod=*/(short)0, acc, /*reuse_a=*/false, /*reuse_b=*/false);
    }

    const int col = ct * 16 + l16;
    const float bv  = bp[col];
    const float bv2 = Agg ? bias2[col] : 0.0f;
#pragma unroll
    for (int j = 0; j < 8; ++j) {
        int m  = rt * 16 + j + half * 8;
        int gr = rowBase + m;
        if (gr < n) {
            float v = acc[j] + bv;
            if (Agg) {
                Out[(size_t)gr * HID + col] = v;               // t for the gather
                float di = dinv[gr];
                Agg[(size_t)gr * HID + col] = bv2 + v * di * di;  // bias + self loop
            } else {
                if (doRelu) v = fmaxf(v, 0.0f);
                Out[(size_t)gr * HID + col] = v;
            }
        }
    }
}

// ---------------------------------------------------------------------------
// Edge scatter: 16 threads/edge, float4 gather, 4 atomic f32 adds each.
// ---------------------------------------------------------------------------
__global__ __launch_bounds__(256) void edge_msgs_kernel(const long long* __restrict__ src,
                                                        const long long* __restrict__ dst,
                                                        const float* __restrict__ wedge,
                                                        const float* __restrict__ t,
                                                        float* __restrict__ agg, int e) {
    long long g   = (long long)blockIdx.x * blockDim.x + threadIdx.x;
    int       eid = (int)(g >> 4);
    int       q   = (int)(g & 15);
    if (eid < e) {
        int    s = (int)src[eid];
        int    d = (int)dst[eid];
        float  w = wedge[eid];
        float4 v = ((const float4*)t)[(size_t)s * 16 + q];
        float* a = agg + (size_t)d * HID + q * 4;
        atomicAdd(a + 0, v.x * w);
        atomicAdd(a + 1, v.y * w);
        atomicAdd(a + 2, v.z * w);
        atomicAdd(a + 3, v.w * w);
    }
}

// ---------------------------------------------------------------------------
// Host-side orchestration
// ---------------------------------------------------------------------------
extern "C" void kernel_launch(void* const* d_in, const int* in_sizes, int n_in,
                              void* d_out, int out_size, void* d_ws, size_t ws_size,
                              hipStream_t stream) {
    const float*     x          = (const float*)d_in[0];
    const long long* edge_index = (const long long*)d_in[1];
    const float*     bn0_gamma  = (const float*)d_in[2];
    const float*     bn0_beta   = (const float*)d_in[3];
    const float*     W0         = (const float*)d_in[4];
    const float*     b0         = (const float*)d_in[5];
    const float*     bns_gamma  = (const float*)d_in[6];
    const float*     bns_beta   = (const float*)d_in[7];
    const float*     Ws         = (const float*)d_in[8];
    const float*     bs         = (const float*)d_in[9];

    const int n = in_sizes[0] / 128;  // N_NODES
    const int E = in_sizes[1] / 2;    // N_EDGES
    const long long* src = edge_index;      // row 0
    const long long* dst = edge_index + E;  // row 1

    float* h = (float*)d_out;  // activation buffer lives in d_out

    // workspace carve-up (256B aligned)
    char*  ws    = (char*)d_ws;
    size_t off   = 0;
    auto   carve = [&](size_t bytes) { void* p = ws + off; off = (off + bytes + 255) & ~(size_t)255; return p; };
    float* dinv  = (float*)carve((size_t)n * 4);
    float* t     = (float*)carve((size_t)n * HID * 4);
    float* agg   = (float*)carve((size_t)n * HID * 4);
    float* wedge = (float*)carve((size_t)E * 4);
    float* stats = (float*)carve(256 * 4);
    float* WpA   = (float*)carve(128 * HID * 4);
    float* bpA   = (float*)carve(HID * 4);
    (void)ws_size;

    const int   TPB     = 256;
    const float invN    = 1.0f / (float)n;
    const int   nBlk    = (n + TPB - 1) / TPB;
    const int   n64Blk  = (n * HID + TPB - 1) / TPB;
    const int   gemmBlk = (n + 31) / 32;
    const int   eBlk    = (E + TPB - 1) / TPB;
    const long long eThreads = (long long)E * 16;
    const int   edgeBlk = (int)((eThreads + TPB - 1) / TPB);

    // --- degree / dinv / per-edge weights (identical across layers) ---
    deg_init_kernel<<<nBlk, TPB, 0, stream>>>(dinv, n);
    deg_edges_kernel<<<eBlk, TPB, 0, stream>>>(dst, dinv, E);
    deg_finish_kernel<<<nBlk, TPB, 0, stream>>>(dinv, n);
    edge_w_kernel<<<eBlk, TPB, 0, stream>>>(src, dst, dinv, wedge, E);

    // --- layer 0: BN(x) folded into W0; h = relu(x @ W0' + b0') ---
    stats_zero_kernel<<<1, 256, 0, stream>>>(stats);
    stats_kernel<<<2048, 256, 0, stream>>>(x, stats, n, 128);
    prep_kernel<<<1, 256, 0, stream>>>(stats, bn0_gamma, bn0_beta, W0, b0, WpA, bpA, 128, invN);
    gemm_wmma_kernel<<<gemmBlk, 256, 0, stream>>>(x, WpA, bpA, h,
                                                  nullptr, nullptr, nullptr, n, 128, /*relu=*/1);

    // stats of h for the first hidden layer's BN
    stats_zero_kernel<<<1, 256, 0, stream>>>(stats);
    stats_kernel<<<2048, 256, 0, stream>>>(h, stats, n, HID);

    // --- stacked BN -> GCNConv -> ReLU layers ---
    for (int layer = 0; layer < 3; ++layer) {
        const float* g  = bns_gamma + layer * HID;
        const float* bt = bns_beta  + layer * HID;
        const float* W  = Ws + (size_t)layer * HID * HID;
        const float* b  = bs + (size_t)layer * HID;

        prep_kernel<<<1, 256, 0, stream>>>(stats, g, bt, W, nullptr, WpA, bpA, HID, invN);
        // t = BN(h) @ W  (BN folded); fused epilogue: agg = b + t*dinv^2
        gemm_wmma_kernel<<<gemmBlk, 256, 0, stream>>>(h, WpA, bpA, t,
                                                      dinv, b, agg, n, HID, /*relu=*/0);
        // scatter messages into agg
        edge_msgs_kernel<<<edgeBlk, TPB, 0, stream>>>(src, dst, wedge, t, agg, E);
        // h = relu(agg); fuse next layer's BN stats when one exists
        if (layer < 2) {
            stats_zero_kernel<<<1, 256, 0, stream>>>(stats);
            relu_stats_kernel<<<2048, 256, 0, stream>>>(agg, h, stats, n);
        } else {
            relu_kernel<<<n64Blk, TPB, 0, stream>>>(agg, h, n * HID);
        }
    }
}